// NLRidge_83983790506106
// MI455X (gfx1250) — compile-verified
//
#include <hip/hip_runtime.h>

typedef float v2f __attribute__((ext_vector_type(2)));
typedef float v8f __attribute__((ext_vector_type(8)));

// Problem constants (N=1, C=3, H=W=256, P1=P2=7, WIN=37, STEP=4, M1=18, M2=55)
constexpr int IMH   = 256;
constexpr int IMW   = 256;
constexpr int NPIX  = IMH * IMW;          // 65536
constexpr int NIMG  = 3 * NPIX;           // 196608
constexpr int PSZ   = 7;
constexpr int HQ    = 250;                // H - P + 1
constexpr int WQ    = 250;
constexpr int WINSZ = 37;
constexpr int RAD   = 18;
constexpr int NOFF  = WINSZ * WINSZ;      // 1369
constexpr int NQ    = 64 * 64;            // 4096 query positions (stride-4 grid + last row/col)
constexpr int NFEAT = 3 * PSZ * PSZ;      // 147
constexpr int WD    = 2 * RAD + PSZ;      // 43 (LDS search window)

// ---------------- mean over channels ----------------
__global__ void k_mean(const float* __restrict__ img, float* __restrict__ out) {
    int p = blockIdx.x * blockDim.x + threadIdx.x;
    if (p < NPIX) out[p] = (img[p] + img[p + NPIX] + img[p + 2 * NPIX]) * (1.0f / 3.0f);
}

__global__ void k_zero(float* __restrict__ a, int n) {
    int i = blockIdx.x * blockDim.x + threadIdx.x;
    if (i < n) a[i] = 0.f;
}

__global__ void k_div(const float* __restrict__ num, const float* __restrict__ den,
                      float* __restrict__ out) {
    int i = blockIdx.x * blockDim.x + threadIdx.x;
    if (i < NIMG) out[i] = num[i] / den[i];
}

// ---------------- block matching: one workgroup per query position ----------------
__global__ void __launch_bounds__(256) k_blockmatch(const float* __restrict__ meanimg,
                                                    int* __restrict__ idxout, int M) {
    __shared__ float win[WD * WD];
    __shared__ float dist[NOFF];
    __shared__ float rv[256];
    __shared__ int   ri[256];
    const int tid = threadIdx.x;
    const int qp  = blockIdx.x;
    const int qy = qp >> 6, qx = qp & 63;
    const int row0 = (qy < 63) ? qy * 4 : 249;   // subsample rows: 0,4,...,248,249
    const int col0 = (qx < 63) ? qx * 4 : 249;

    for (int t = tid; t < WD * WD; t += 256) {
        int wr = t / WD, wc = t % WD;
        int gr = row0 - RAD + wr, gc = col0 - RAD + wc;
        win[t] = (gr >= 0 && gr < IMH && gc >= 0 && gc < IMW) ? meanimg[gr * IMW + gc] : 0.f;
    }
    __syncthreads();

    for (int off = tid; off < NOFF; off += 256) {
        int di = off / WINSZ, dj = off % WINSZ;
        int a = row0 + di - RAD, b = col0 + dj - RAD;
        float d;
        if (a >= 0 && a < HQ && b >= 0 && b < WQ) {
            d = 0.f;
            #pragma unroll
            for (int i = 0; i < PSZ; ++i)
                #pragma unroll
                for (int j = 0; j < PSZ; ++j) {
                    float t0 = win[(di + i) * WD + dj + j] - win[(RAD + i) * WD + RAD + j];
                    d += t0 * t0;
                }
        } else {
            d = 1e30f;
        }
        if (off == RAD * WINSZ + RAD) d = -1e10f;   // center always selected first
        dist[off] = d;
    }
    __syncthreads();

    // iterative top-M (smallest distance, lowest-index tie-break)
    for (int k = 0; k < M; ++k) {
        float bv = 3.0e38f; int bi = 0x7fffffff;
        for (int off = tid; off < NOFF; off += 256) {
            float d = dist[off];
            if (d < bv) { bv = d; bi = off; }
        }
        rv[tid] = bv; ri[tid] = bi;
        __syncthreads();
        for (int s = 128; s > 0; s >>= 1) {
            if (tid < s) {
                float ov = rv[tid + s]; int oi = ri[tid + s];
                if (ov < rv[tid] || (ov == rv[tid] && oi < ri[tid])) { rv[tid] = ov; ri[tid] = oi; }
            }
            __syncthreads();
        }
        if (tid == 0) {
            int best = ri[0];
            int di = best / WINSZ - RAD, dj = best % WINSZ - RAD;
            idxout[qp * M + k] = (row0 + di) * WQ + (col0 + dj);
            dist[best] = 3.0e38f;
        }
        __syncthreads();
    }
}

// ---------------- per-group ridge denoise (FP32 WMMA) ----------------
// theta = I - n*sigma^2 * inv(A),  A = Gram(gsrc-patches) (+ n*sigma^2*I for stage 2)
// X_hat = theta @ Y(ysrc-patches);  scatter X_hat*w and w with atomics.
template <int M, int MP, bool ADD_DIAG>
__global__ void __launch_bounds__(256) k_denoise(const float* __restrict__ gsrc,
                                                 const float* __restrict__ ysrc,
                                                 const int* __restrict__ idx,
                                                 const float* __restrict__ sigma,
                                                 float* __restrict__ num,
                                                 float* __restrict__ den) {
    constexpr int FSTR = 160;       // padded feature stride (>= 10*16)
    constexpr int GSTR = MP + 2;
    constexpr int MT   = MP / 16;
    __shared__ float patch[MP * FSTR];   // zero-padded M x 147 patch matrix
    __shared__ float gram[MP * GSTR];    // Gram -> inverse -> theta (pad stays 0)
    __shared__ float colk[MP];
    __shared__ float wts[MP];
    __shared__ int   pids[MP];
    __shared__ float pivinv_s;

    const int tid  = threadIdx.x;
    const int lane = tid & 31;
    const int wid  = tid >> 5;
    const int g    = blockIdx.x;
    const float sg  = sigma[0];
    const float ns2 = (float)NFEAT * sg * sg;

    for (int t = tid; t < MP * FSTR; t += 256) patch[t] = 0.f;
    if (tid < MP) pids[tid] = (tid < M) ? idx[g * M + tid] : 0;
    __syncthreads();

    // gather patches for the Gram source
    for (int t = tid; t < M * NFEAT; t += 256) {
        int row = t / NFEAT, f = t % NFEAT;
        int pid = pids[row];
        int r = pid / WQ, q = pid % WQ;
        int c = f / 49, rem = f % 49;
        patch[row * FSTR + f] = gsrc[(c * IMH + r + rem / 7) * IMW + q + rem % 7];
    }
    __syncthreads();

    // Gram = P * P^T via v_wmma_f32_16x16x4_f32 (one 16x16 tile per wave)
    {
        const int krow = lane & 15;               // M row of A / N col of B
        const int koff = (lane >> 4) << 1;        // K pair: lanes 0-15 -> K=0,1; 16-31 -> K=2,3
        for (int tile = wid; tile < MT * MT; tile += 8) {
            int ti = tile / MT, tj = tile % MT;
            v8f acc = {0.f, 0.f, 0.f, 0.f, 0.f, 0.f, 0.f, 0.f};
            for (int f = 0; f < 148; f += 4) {
                v2f a, b;
                a.x = patch[(ti * 16 + krow) * FSTR + f + koff];
                a.y = patch[(ti * 16 + krow) * FSTR + f + koff + 1];
                b.x = patch[(tj * 16 + krow) * FSTR + f + koff];   // B[k][n] = P[n][k]
                b.y = patch[(tj * 16 + krow) * FSTR + f + koff + 1];
                acc = __builtin_amdgcn_wmma_f32_16x16x4_f32(false, a, false, b,
                                                            (short)0, acc, false, false);
            }
            #pragma unroll
            for (int v = 0; v < 8; ++v)   // C/D layout: row = v + 8*lane[4], col = lane&15
                gram[(ti * 16 + v + ((lane >> 4) << 3)) * GSTR + tj * 16 + krow] = acc[v];
        }
    }
    __syncthreads();

    if (ADD_DIAG) {
        if (tid < M) gram[tid * GSTR + tid] += ns2;
        __syncthreads();
    }

    // in-place Gauss-Jordan inversion of top-left MxM (SPD, no pivoting)
    for (int k = 0; k < M; ++k) {
        if (tid < M) colk[tid] = gram[tid * GSTR + k];
        if (tid == 0) pivinv_s = 1.0f / gram[k * GSTR + k];
        __syncthreads();
        const float pinv = pivinv_s;
        for (int j = tid; j < M; j += 256)
            gram[k * GSTR + j] = (j == k) ? pinv : gram[k * GSTR + j] * pinv;
        __syncthreads();
        for (int t = tid; t < M * M; t += 256) {
            int i = t / M, j = t % M;
            if (i != k) {
                float f0 = colk[i];
                gram[i * GSTR + j] = (j == k) ? (-f0 * pinv)
                                              : (gram[i * GSTR + j] - f0 * gram[k * GSTR + j]);
            }
        }
        __syncthreads();
    }

    // theta = I - ns2 * inv (in place; pad region already zero)
    for (int t = tid; t < M * M; t += 256) {
        int i = t / M, j = t % M;
        gram[i * GSTR + j] = ((i == j) ? 1.0f : 0.0f) - ns2 * gram[i * GSTR + j];
    }
    __syncthreads();

    // weights = 1 / rowsum(theta^2)
    if (tid < M) {
        float s = 0.f;
        for (int j = 0; j < M; ++j) { float th = gram[tid * GSTR + j]; s += th * th; }
        wts[tid] = 1.0f / s;
    }
    __syncthreads();

    // regather Y (noisy patches) into the same LDS buffer
    for (int t = tid; t < M * NFEAT; t += 256) {
        int row = t / NFEAT, f = t % NFEAT;
        int pid = pids[row];
        int r = pid / WQ, q = pid % WQ;
        int c = f / 49, rem = f % 49;
        patch[row * FSTR + f] = ysrc[(c * IMH + r + rem / 7) * IMW + q + rem % 7];
    }
    __syncthreads();

    // X_hat = theta @ Y via WMMA, scatter-accumulate weighted patches
    {
        const int krow = lane & 15;
        const int koff = (lane >> 4) << 1;
        for (int tile = wid; tile < MT * 10; tile += 8) {
            int ti = tile / 10, fj = tile % 10;
            v8f acc = {0.f, 0.f, 0.f, 0.f, 0.f, 0.f, 0.f, 0.f};
            for (int k = 0; k < MP; k += 4) {
                v2f a, b;
                a.x = gram[(ti * 16 + krow) * GSTR + k + koff];
                a.y = gram[(ti * 16 + krow) * GSTR + k + koff + 1];
                b.x = patch[(k + koff) * FSTR + fj * 16 + krow];
                b.y = patch[(k + koff + 1) * FSTR + fj * 16 + krow];
                acc = __builtin_amdgcn_wmma_f32_16x16x4_f32(false, a, false, b,
                                                            (short)0, acc, false, false);
            }
            #pragma unroll
            for (int v = 0; v < 8; ++v) {
                int row = ti * 16 + v + ((lane >> 4) << 3);
                int col = fj * 16 + krow;
                if (row < M && col < NFEAT) {
                    float wv  = wts[row];
                    int   pid = pids[row];
                    int r = pid / WQ, q = pid % WQ;
                    int c = col / 49, rem = col % 49;
                    int addr = (c * IMH + r + rem / 7) * IMW + q + rem % 7;
                    atomicAdd(&num[addr], acc[v] * wv);
                    atomicAdd(&den[addr], wv);
                }
            }
        }
    }
}

// ---------------- host-side pipeline ----------------
extern "C" void kernel_launch(void* const* d_in, const int* in_sizes, int n_in,
                              void* d_out, int out_size, void* d_ws, size_t ws_size,
                              hipStream_t stream) {
    (void)in_sizes; (void)n_in; (void)out_size; (void)ws_size;
    const float* y     = (const float*)d_in[0];
    const float* sigma = (const float*)d_in[1];
    float* out = (float*)d_out;

    float* ws   = (float*)d_ws;
    float* mean = ws;                       // 65536 floats
    float* num  = ws + NPIX;                // 196608
    float* den  = num + NIMG;               // 196608 (contiguous with num)
    float* den1 = den + NIMG;               // 196608
    int*   idx  = (int*)(den1 + NIMG);      // 4096*55 ints

    // ---- stage 1: M1=18 ----
    k_mean<<<NPIX / 256, 256, 0, stream>>>(y, mean);
    k_blockmatch<<<NQ, 256, 0, stream>>>(mean, idx, 18);
    k_zero<<<(2 * NIMG + 255) / 256, 256, 0, stream>>>(num, 2 * NIMG);
    k_denoise<18, 32, false><<<NQ, 256, 0, stream>>>(y, y, idx, sigma, num, den);
    k_div<<<(NIMG + 255) / 256, 256, 0, stream>>>(num, den, den1);

    // ---- stage 2: M2=55 ----
    k_mean<<<NPIX / 256, 256, 0, stream>>>(den1, mean);
    k_blockmatch<<<NQ, 256, 0, stream>>>(mean, idx, 55);
    k_zero<<<(2 * NIMG + 255) / 256, 256, 0, stream>>>(num, 2 * NIMG);
    k_denoise<55, 64, true><<<NQ, 256, 0, stream>>>(den1, y, idx, sigma, num, den);
    k_div<<<(NIMG + 255) / 256, 256, 0, stream>>>(num, den, out);
}